// MultiHeadAttention_75737453297867
// MI455X (gfx1250) — compile-verified
//
#include <hip/hip_runtime.h>
#include <math.h>

// ---------------------------------------------------------------------------
// Local 2D attention (bug-compatible as_strided flange gather) for MI455X.
// Shapes (static): B=1, C_IN=64, C_OUT=64, H=128, W=48, D_MODEL=32, N_HEADS=8,
// DPH=4, Q0=128, Q1=24, F0=F1=8  ->  Hp=128, Wp=48, nH=1, nW=2, M0=144, M1=40.
// Attention blocks: N = 8 heads * 2 width-blocks = 16; q_len=3072, kv_len=5760.
// QK^T uses V_WMMA_F32_16X16X4_F32 (16x4 A tile == 16 queries x dph=4, full
// f32 so it matches the f32 reference). Softmax is streamed per-lane (flash
// style) with a single-exp update (exp(-|d|) + cndmask selects), merged once
// at the end via shfl_xor. Loop is VALU/exp bound; K/V live in L2 (192 MB).
// ---------------------------------------------------------------------------

typedef __attribute__((ext_vector_type(2))) float v2f;
typedef __attribute__((ext_vector_type(8))) float v8f;

#define CH_SP   6144      // 128*48 spatial elements per channel
#define KP_CH   9216      // 144*64 padded spatial per channel
#define KP_TOT  294912    // 32 * 9216

// ------------------------- workspace zero-init ------------------------------
__global__ void zero_ws_kernel(float* __restrict__ p, int n) {
    int i = blockIdx.x * blockDim.x + threadIdx.x;
    if (i < n) p[i] = 0.0f;
}

// ------------------- fused q/k/v 3x3 convolutions ---------------------------
// t==0 -> q (scaled by dph^-0.5 = 0.5) into qbuf[32][128][48]
// t==1 -> k into kpad[8][4][144][64] interior (flange stays zero)
// t==2 -> v into vpad likewise
__global__ void conv_qkv_kernel(const float* __restrict__ x,
                                const float* __restrict__ wq, const float* __restrict__ bq,
                                const float* __restrict__ wk, const float* __restrict__ bk,
                                const float* __restrict__ wv, const float* __restrict__ bv,
                                float* __restrict__ qbuf,
                                float* __restrict__ kpad,
                                float* __restrict__ vpad) {
    int idx = blockIdx.x * blockDim.x + threadIdx.x;
    if (idx >= 3 * 32 * CH_SP) return;
    int t   = idx / (32 * CH_SP);
    int rem = idx - t * (32 * CH_SP);
    int co  = rem / CH_SP;
    int sp  = rem - co * CH_SP;
    int y   = sp / 48;
    int xx  = sp - y * 48;

    const float* w; const float* bias;
    if (t == 0)      { w = wq; bias = bq; }
    else if (t == 1) { w = wk; bias = bk; }
    else             { w = wv; bias = bv; }

    float acc = bias[co];
    for (int ci = 0; ci < 64; ++ci) {
        const float* xp = x + ci * CH_SP;
        const float* wp = w + (co * 64 + ci) * 9;
#pragma unroll
        for (int ky = 0; ky < 3; ++ky) {
            int yy = y + ky - 1;
            if (yy < 0 || yy >= 128) continue;
#pragma unroll
            for (int kx = 0; kx < 3; ++kx) {
                int xc = xx + kx - 1;
                if (xc < 0 || xc >= 48) continue;
                acc = fmaf(xp[yy * 48 + xc], wp[ky * 3 + kx], acc);
            }
        }
    }
    if (t == 0) {
        qbuf[rem] = acc * 0.5f;                       // dph^-0.5
    } else {
        float* dst = (t == 1) ? kpad : vpad;
        dst[co * KP_CH + (y + 8) * 64 + (xx + 8)] = acc;
    }
}

// --------------------------- attention core ---------------------------------
// grid = (192 query tiles, 8 heads, 2 width-blocks), block = 32 (one wave).
// Each wave: 16-query tile, streams 360 chunks of 16 KV columns.
__global__ __launch_bounds__(32)
void attn_wmma_kernel(const float* __restrict__ qbuf,
                      const float* __restrict__ kpad,
                      const float* __restrict__ vpad,
                      float* __restrict__ obuf) {
    const int lane  = threadIdx.x;       // 0..31
    const int tile  = blockIdx.x;        // 0..191
    const int h     = blockIdx.y;        // head 0..7
    const int j     = blockIdx.z;        // width block 0..1
    const int mrow  = lane & 15;         // A-matrix M (query row in tile)
    const int n     = lane & 15;         // B/C/D N (kv column within chunk)
    const int khalf = (lane >> 4) << 1;  // K-pair this lane holds: 0 or 2

    // ---- A operand: 16 queries x dph=4 (per ISA 16x4 f32 layout) ----
    const int q    = tile * 16 + mrow;            // 0..3071
    const int q0   = q / 24;
    const int q1   = q - q0 * 24;
    const int qsp  = q0 * 48 + j * 24 + q1;       // spatial offset in [128][48]
    v2f a;
    a.x = qbuf[(h * 4 + khalf)     * CH_SP + qsp];
    a.y = qbuf[(h * 4 + khalf + 1) * CH_SP + qsp];

    // bug-compatible as_strided flat base for this (head, width-block)
    const int baseK = h * 24576 + j * 24;         // + d*6144 + m0*48 + m1

    float mr[8], lr[8], acc[8][4];
#pragma unroll
    for (int r = 0; r < 8; ++r) {
        mr[r] = -INFINITY; lr[r] = 0.0f;
#pragma unroll
        for (int d = 0; d < 4; ++d) acc[r][d] = 0.0f;
    }

    for (int chunk = 0; chunk < 360; ++chunk) {   // 360*16 = 5760 kv columns
        const int kv = (chunk << 4) + n;
        const int m0 = kv / 40;
        const int m1 = kv - m0 * 40;
        const int goff = baseK + m0 * 48 + m1;

        // B operand: dph x 16 kv columns (per ISA 4x16 f32 layout)
        v2f b;
        b.x = kpad[goff + khalf * CH_SP];
        b.y = kpad[goff + (khalf + 1) * CH_SP];
        // V for this lane's kv column (4 depth values, stride 6144)
        float v0 = vpad[goff];
        float v1 = vpad[goff + 1 * CH_SP];
        float v2 = vpad[goff + 2 * CH_SP];
        float v3 = vpad[goff + 3 * CH_SP];

        v8f c = {0.f, 0.f, 0.f, 0.f, 0.f, 0.f, 0.f, 0.f};
        c = __builtin_amdgcn_wmma_f32_16x16x4_f32(
                /*neg_a=*/false, a, /*neg_b=*/false, b,
                /*c_mod=*/(short)0, c, /*reuse_a=*/false, /*reuse_b=*/false);

        // per-lane online softmax over this lane's kv-column subset;
        // C/D layout: VGPR r holds rows r (lanes 0-15) and r+8 (lanes 16-31).
        // Single-exp update: one of exp(s-mn), exp(m_old-mn) is always 1, so
        // compute e = exp(-|s - m_old|) once and select p/sc with cndmask.
#pragma unroll
        for (int r = 0; r < 8; ++r) {
            float s  = c[r];
            float d  = s - mr[r];
            bool  g  = d > 0.0f;                  // new max?
            float e  = __expf(g ? -d : d);        // exp(-|d|)  (1 v_exp_f32)
            float p  = g ? 1.0f : e;              // exp(s - mn)
            float sc = g ? e : 1.0f;              // exp(m_old - mn)
            mr[r] = g ? s : mr[r];
            lr[r] = fmaf(lr[r], sc, p);
            acc[r][0] = fmaf(p, v0, acc[r][0] * sc);
            acc[r][1] = fmaf(p, v1, acc[r][1] * sc);
            acc[r][2] = fmaf(p, v2, acc[r][2] * sc);
            acc[r][3] = fmaf(p, v3, acc[r][3] * sc);
        }
    }

    // ---- merge the 16 column-partials (stays inside each lane half) ----
#pragma unroll
    for (int off = 1; off < 16; off <<= 1) {
#pragma unroll
        for (int r = 0; r < 8; ++r) {
            float mo = __shfl_xor(mr[r],     off, 32);
            float lo = __shfl_xor(lr[r],     off, 32);
            float a0 = __shfl_xor(acc[r][0], off, 32);
            float a1 = __shfl_xor(acc[r][1], off, 32);
            float a2 = __shfl_xor(acc[r][2], off, 32);
            float a3 = __shfl_xor(acc[r][3], off, 32);
            float d  = mo - mr[r];
            bool  g  = d > 0.0f;                  // other partial has the max
            float e  = __expf(g ? -d : d);        // exp(-|d|)
            float e1 = g ? e : 1.0f;              // scale for self
            float e2 = g ? 1.0f : e;              // scale for other
            mr[r] = g ? mo : mr[r];
            lr[r] = lr[r] * e1 + lo * e2;
            acc[r][0] = acc[r][0] * e1 + a0 * e2;
            acc[r][1] = acc[r][1] * e1 + a1 * e2;
            acc[r][2] = acc[r][2] * e1 + a2 * e2;
            acc[r][3] = acc[r][3] * e1 + a3 * e2;
        }
    }

    // lane 0 writes rows 0..7 of tile, lane 16 writes rows 8..15
    if (n == 0) {
#pragma unroll
        for (int r = 0; r < 8; ++r) {
            int row = tile * 16 + r + ((lane >> 4) << 3);
            int r0  = row / 24;
            int r1  = row - r0 * 24;
            int sp  = r0 * 48 + j * 24 + r1;
            float inv = 1.0f / lr[r];
#pragma unroll
            for (int d = 0; d < 4; ++d)
                obuf[(h * 4 + d) * CH_SP + sp] = acc[r][d] * inv;
        }
    }
}

// --------------------------- output 3x3 conv --------------------------------
__global__ void conv_out_kernel(const float* __restrict__ obuf,
                                const float* __restrict__ wo,
                                float* __restrict__ out) {
    int idx = blockIdx.x * blockDim.x + threadIdx.x;
    if (idx >= 64 * CH_SP) return;
    int co = idx / CH_SP;
    int sp = idx - co * CH_SP;
    int y  = sp / 48;
    int xx = sp - y * 48;

    float acc = 0.0f;
    for (int ci = 0; ci < 32; ++ci) {
        const float* op = obuf + ci * CH_SP;
        const float* wp = wo + (co * 32 + ci) * 9;
#pragma unroll
        for (int ky = 0; ky < 3; ++ky) {
            int yy = y + ky - 1;
            if (yy < 0 || yy >= 128) continue;
#pragma unroll
            for (int kx = 0; kx < 3; ++kx) {
                int xc = xx + kx - 1;
                if (xc < 0 || xc >= 48) continue;
                acc = fmaf(op[yy * 48 + xc], wp[ky * 3 + kx], acc);
            }
        }
    }
    out[idx] = acc;
}

// ---------------------------------------------------------------------------
extern "C" void kernel_launch(void* const* d_in, const int* in_sizes, int n_in,
                              void* d_out, int out_size, void* d_ws, size_t ws_size,
                              hipStream_t stream) {
    const float* x  = (const float*)d_in[0];
    const float* wq = (const float*)d_in[1];
    const float* bq = (const float*)d_in[2];
    const float* wk = (const float*)d_in[3];
    const float* bk = (const float*)d_in[4];
    const float* wv = (const float*)d_in[5];
    const float* bv = (const float*)d_in[6];
    const float* wo = (const float*)d_in[7];
    float* out = (float*)d_out;

    float* ws   = (float*)d_ws;
    float* qbuf = ws;                         // 196608 floats
    float* kpad = ws + 196608;                // 294912 floats (padded K)
    float* vpad = ws + 196608 + KP_TOT;       // 294912 floats (padded V)
    float* obuf = ws + 196608 + 2 * KP_TOT;   // 196608 floats

    // 1) zero the flange-padded K/V storage (ws is poisoned by harness)
    {
        int n = 2 * KP_TOT;
        zero_ws_kernel<<<(n + 255) / 256, 256, 0, stream>>>(kpad, n);
    }
    // 2) q/k/v convolutions (q pre-scaled by dph^-0.5)
    {
        int n = 3 * 32 * CH_SP;
        conv_qkv_kernel<<<(n + 255) / 256, 256, 0, stream>>>(
            x, wq, bq, wk, bk, wv, bv, qbuf, kpad, vpad);
    }
    // 3) blocked local attention with WMMA f32 16x16x4 + streaming softmax
    attn_wmma_kernel<<<dim3(192, 8, 2), 32, 0, stream>>>(qbuf, kpad, vpad, obuf);
    // 4) output convolution
    {
        int n = 64 * CH_SP;
        conv_out_kernel<<<(n + 255) / 256, 256, 0, stream>>>(obuf, wo, out);
    }
}